// EvoformerIteration_26242250179314
// MI455X (gfx1250) — compile-verified
//
#include <hip/hip_runtime.h>

// ---------------------------------------------------------------------------
// Evoformer iteration for gfx1250 (MI455X), fp32 via V_WMMA_F32_16X16X4_F32.
// Fragment layout assumptions (CDNA5 ISA 7.12.2):
//   A (16x4 f32, v2f): lane<16: M=lane, K={k0,k0+1}; lane>=16: M=lane-16, K={k0+2,k0+3}
//   B (4x16 f32, v2f): mirrored: lane holds N=lane%16, K pair selected by lane/16
//   C/D (16x16 f32, v8f): lane<16: reg r -> (M=r, N=lane); lane>=16: (M=r+8, N=lane-16)
// ---------------------------------------------------------------------------

typedef float v2f __attribute__((ext_vector_type(2)));
typedef float v8f __attribute__((ext_vector_type(8)));

__device__ __forceinline__ v8f wmma4(v2f a, v2f b, v8f c) {
  return __builtin_amdgcn_wmma_f32_16x16x4_f32(false, a, false, b, (short)0, c,
                                               false, false);
}
__device__ __forceinline__ float sigm(float x) { return 1.f / (1.f + __expf(-x)); }
__device__ __forceinline__ v2f ldv2(const float* p) { return *(const v2f*)p; }

#define FLAG_RELU     1
#define FLAG_ACC      4
#define FLAG_MULSIG   8
#define FLAG_ROWMASK 16
#define FLAG_ROWSWAP 32

// ---------------------------------------------------------------------------
// Generic GEMM: out[M,N] (op)= epilogue(A[M,K] @ W[K,N] + bias)
// Block: 256 threads = 8 waves; wave w -> rows [bx*128+16w, +16), cols [by*64, +64)
// NT = number of active 16-col tiles for this wave (branch-free inner loop).
// ---------------------------------------------------------------------------
template <int NT>
__device__ __forceinline__ void gemm_tile(
    const float* __restrict__ A, const float* __restrict__ W,
    const float* __restrict__ bias, float* __restrict__ out,
    const float* __restrict__ aux, int N, int K, int flags, int pmA, int pmB,
    int m0, int nb, int lr, int lh) {
  v8f z = {0.f, 0.f, 0.f, 0.f, 0.f, 0.f, 0.f, 0.f};
  v8f acc[NT];
#pragma unroll
  for (int t = 0; t < NT; t++) acc[t] = z;
  const float* ap0 = A + (size_t)(m0 + lr) * K + lh * 2;
  const float* wp0 = W + (size_t)(lh * 2) * N + nb + lr;
  for (int k0 = 0; k0 < K; k0 += 8) {  // K % 8 == 0 for all call sites
    v2f a0 = ldv2(ap0 + k0);
    v2f a1 = ldv2(ap0 + k0 + 4);
    const float* wp = wp0 + (size_t)k0 * N;
    const float* wq = wp + (size_t)4 * N;
    v2f b0[NT], b1[NT];
#pragma unroll
    for (int t = 0; t < NT; t++) { b0[t].x = wp[t * 16]; b0[t].y = wp[N + t * 16]; }
#pragma unroll
    for (int t = 0; t < NT; t++) { b1[t].x = wq[t * 16]; b1[t].y = wq[N + t * 16]; }
#pragma unroll
    for (int t = 0; t < NT; t++) acc[t] = wmma4(a0, b0[t], acc[t]);
#pragma unroll
    for (int t = 0; t < NT; t++) acc[t] = wmma4(a1, b1[t], acc[t]);
  }
#pragma unroll
  for (int t = 0; t < NT; t++) {
    int nc = nb + t * 16;
#pragma unroll
    for (int r = 0; r < 8; r++) {
      int row = m0 + r + lh * 8;
      int col = nc + lr;
      float vv = acc[t][r];
      if (bias) vv += bias[col];
      if (flags & FLAG_RELU) vv = fmaxf(vv, 0.f);
      if (flags & FLAG_MULSIG) vv *= sigm(aux[(size_t)row * N + col]);
      if (flags & FLAG_ROWMASK) vv *= aux[row];
      int orow = row;
      if (flags & FLAG_ROWSWAP) orow = (row % pmB) * pmA + (row / pmB);
      size_t oi = (size_t)orow * N + col;
      if (flags & FLAG_ACC) out[oi] += vv; else out[oi] = vv;
    }
  }
}

__global__ __launch_bounds__(256) void gemm_wmma(
    const float* __restrict__ A, const float* __restrict__ W,
    const float* __restrict__ bias, float* __restrict__ out,
    const float* __restrict__ aux, int M, int N, int K, int flags, int pmA,
    int pmB) {
  int wave = threadIdx.x >> 5, lane = threadIdx.x & 31;
  int lr = lane & 15, lh = lane >> 4;
  int m0 = blockIdx.x * 128 + wave * 16;
  if (m0 >= M) return;
  int nb = blockIdx.y * 64;
  int rem = N - nb;
  if (rem >= 64)
    gemm_tile<4>(A, W, bias, out, aux, N, K, flags, pmA, pmB, m0, nb, lr, lh);
  else if (rem >= 32)
    gemm_tile<2>(A, W, bias, out, aux, N, K, flags, pmA, pmB, m0, nb, lr, lh);
  else
    gemm_tile<1>(A, W, bias, out, aux, N, K, flags, pmA, pmB, m0, nb, lr, lh);
}

// Small-N GEMM with transposed store: out[n*M + m] = A[m,:] . W[:,n]  (pair-bias proj)
__global__ __launch_bounds__(256) void gemm_small_n_t(
    const float* __restrict__ A, const float* __restrict__ W,
    float* __restrict__ out, int M, int N, int K) {
  int m = blockIdx.x * 256 + threadIdx.x;
  int n = blockIdx.y;
  if (m >= M) return;
  const float* ap = A + (size_t)m * K;
  float s = 0.f;
  for (int kk = 0; kk < K; kk++) s += ap[kk] * W[(size_t)kk * N + n];
  out[(size_t)n * M + m] = s;
}

// ---------------------------------------------------------------------------
// LayerNorm over last dim (one wave per row)
// ---------------------------------------------------------------------------
__global__ __launch_bounds__(256) void ln_kernel(
    const float* __restrict__ x, const float* __restrict__ g,
    const float* __restrict__ b, float* __restrict__ out, int R, int C) {
  int wave = threadIdx.x >> 5, lane = threadIdx.x & 31;
  int row = blockIdx.x * 8 + wave;
  if (row >= R) return;
  const float* xp = x + (size_t)row * C;
  float s = 0.f, ss = 0.f;
  for (int i = lane; i < C; i += 32) { float v = xp[i]; s += v; ss += v * v; }
  for (int off = 16; off >= 1; off >>= 1) {
    s += __shfl_xor(s, off, 32);
    ss += __shfl_xor(ss, off, 32);
  }
  float mu = s / C;
  float var = ss / C - mu * mu;
  float rs = rsqrtf(var + 1e-5f);
  float* op = out + (size_t)row * C;
  for (int i = lane; i < C; i += 32) op[i] = (xp[i] - mu) * rs * g[i] + b[i];
}

// ---------------------------------------------------------------------------
// Flash attention (gated MHA, head dim 32). One block per (group, head).
// q/k/v/g: [G, N, H*32]. pb: [H, N, N] or null. kb: [G, N] or null.
// ---------------------------------------------------------------------------
__global__ __launch_bounds__(256) void attn_kernel(
    const float* __restrict__ q, const float* __restrict__ k,
    const float* __restrict__ v, const float* __restrict__ g,
    float* __restrict__ out, const float* __restrict__ pb,
    const float* __restrict__ kb, int N, int H, float scale) {
  __shared__ float Vsh[256 * 32];  // 32 KB
  __shared__ float Ssh[8 * 256];   // 8 KB (per-wave 16x16 P tile)
  int gi = blockIdx.x, h = blockIdx.y;
  int HS = H * 32;
  size_t base = ((size_t)gi * N) * HS + h * 32;
  for (int t = threadIdx.x; t < N * 32; t += 256) {
    int n = t >> 5, c = t & 31;
    Vsh[t] = v[base + (size_t)n * HS + c];
  }
  __syncthreads();
  int wave = threadIdx.x >> 5, lane = threadIdx.x & 31;
  int lr = lane & 15, lh = lane >> 4;
  float* ss = &Ssh[wave * 256];
  int ntiles = N >> 4;
  for (int qt = wave; qt < ntiles; qt += 8) {
    int m0 = qt << 4;
    const float* qp = q + base + (size_t)(m0 + lr) * HS + lh * 2;
    v2f qa[8];
#pragma unroll
    for (int kk = 0; kk < 8; kk++) qa[kk] = ldv2(qp + kk * 4);
    v8f z = {0.f, 0.f, 0.f, 0.f, 0.f, 0.f, 0.f, 0.f};
    v8f o0 = z, o1 = z;
    float mr[8], lsum[8];
#pragma unroll
    for (int r = 0; r < 8; r++) { mr[r] = -1e30f; lsum[r] = 0.f; }
    for (int j0 = 0; j0 < N; j0 += 16) {
      v8f s = z;
      const float* kp = k + base + (size_t)(j0 + lr) * HS + lh * 2;
#pragma unroll
      for (int kk = 0; kk < 8; kk++) {
        v2f bb = ldv2(kp + kk * 4);
        s = wmma4(qa[kk], bb, s);
      }
#pragma unroll
      for (int r = 0; r < 8; r++) {
        int row = m0 + r + lh * 8;
        int col = j0 + lr;
        float val = s[r] * scale;
        if (pb) val += pb[((size_t)h * N + row) * N + col];
        if (kb) val += kb[(size_t)gi * N + col];
        float mx = val;
        mx = fmaxf(mx, __shfl_xor(mx, 1, 32));
        mx = fmaxf(mx, __shfl_xor(mx, 2, 32));
        mx = fmaxf(mx, __shfl_xor(mx, 4, 32));
        mx = fmaxf(mx, __shfl_xor(mx, 8, 32));
        float mnew = fmaxf(mr[r], mx);
        float corr = __expf(mr[r] - mnew);
        float p = __expf(val - mnew);
        float ps = p;
        ps += __shfl_xor(ps, 1, 32);
        ps += __shfl_xor(ps, 2, 32);
        ps += __shfl_xor(ps, 4, 32);
        ps += __shfl_xor(ps, 8, 32);
        lsum[r] = lsum[r] * corr + ps;
        mr[r] = mnew;
        o0[r] *= corr; o1[r] *= corr;
        ss[(r + lh * 8) * 16 + lr] = p;
      }
#pragma unroll
      for (int kk = 0; kk < 4; kk++) {
        int ka = kk * 4 + lh * 2;
        v2f a2 = ldv2(ss + lr * 16 + ka);
        v2f b0; b0.x = Vsh[(j0 + ka) * 32 + lr];
                b0.y = Vsh[(j0 + ka + 1) * 32 + lr];
        v2f b1; b1.x = Vsh[(j0 + ka) * 32 + 16 + lr];
                b1.y = Vsh[(j0 + ka + 1) * 32 + 16 + lr];
        o0 = wmma4(a2, b0, o0);
        o1 = wmma4(a2, b1, o1);
      }
    }
#pragma unroll
    for (int r = 0; r < 8; r++) {
      int row = m0 + r + lh * 8;
      float inv = 1.f / lsum[r];
      size_t ro = base + (size_t)row * HS;
      out[ro + lr]      = o0[r] * inv * sigm(g[ro + lr]);
      out[ro + 16 + lr] = o1[r] * inv * sigm(g[ro + 16 + lr]);
    }
  }
}

// ---------------------------------------------------------------------------
// Fused transition: out[m,:] += W2 @ relu(W1 @ xin[m,:] + b1) + b2
// ---------------------------------------------------------------------------
template <int C, int HD>
__global__ __launch_bounds__(256) void transition_kernel(
    const float* __restrict__ xin, const float* __restrict__ W1,
    const float* __restrict__ b1, const float* __restrict__ W2,
    const float* __restrict__ b2, float* __restrict__ out) {
  __shared__ float hid[16 * HD];
  int wave = threadIdx.x >> 5, lane = threadIdx.x & 31;
  int lr = lane & 15, lh = lane >> 4;
  int m0 = blockIdx.x * 16;
  v8f z = {0.f, 0.f, 0.f, 0.f, 0.f, 0.f, 0.f, 0.f};
  const float* ap0 = xin + (size_t)(m0 + lr) * C + lh * 2;
  for (int nt = wave; nt < (HD >> 4); nt += 8) {
    v8f acc = z;
    const float* wp0 = W1 + (size_t)(lh * 2) * HD + nt * 16 + lr;
    for (int k0 = 0; k0 < C; k0 += 4) {
      v2f a = ldv2(ap0 + k0);
      const float* wp = wp0 + (size_t)k0 * HD;
      v2f b; b.x = wp[0]; b.y = wp[HD];
      acc = wmma4(a, b, acc);
    }
#pragma unroll
    for (int r = 0; r < 8; r++) {
      int rw = r + lh * 8, col = nt * 16 + lr;
      hid[rw * HD + col] = fmaxf(acc[r] + b1[col], 0.f);
    }
  }
  __syncthreads();
  for (int nt = wave; nt < (C >> 4); nt += 8) {
    v8f acc = z;
    const float* wp0 = W2 + (size_t)(lh * 2) * C + nt * 16 + lr;
    for (int k0 = 0; k0 < HD; k0 += 4) {
      v2f a = ldv2(&hid[lr * HD + k0 + lh * 2]);
      const float* wp = wp0 + (size_t)k0 * C;
      v2f b; b.x = wp[0]; b.y = wp[C];
      acc = wmma4(a, b, acc);
    }
#pragma unroll
    for (int r = 0; r < 8; r++) {
      int row = m0 + r + lh * 8, col = nt * 16 + lr;
      out[(size_t)row * C + col] += acc[r] + b2[col];
    }
  }
}

// ---------------------------------------------------------------------------
// Outer product mean, fused: pair[i0+i, j, :] += (P_ij @ Wf + bf) / (norm+1e-3)
// a,b: [64, 256, 32] (m, res, mid) already masked. Block = (i-tile 16, j).
// ---------------------------------------------------------------------------
__global__ __launch_bounds__(256) void opm_kernel(
    const float* __restrict__ a, const float* __restrict__ b,
    const float* __restrict__ Wf, const float* __restrict__ bf,
    const float* __restrict__ mask, float* __restrict__ pair) {
  __shared__ float bcol[64 * 32];  // 8 KB
  __shared__ float P[16 * 512];    // 32 KB (half of the 16x1024 outer matrix)
  __shared__ float nrm[16];
  int i0 = blockIdx.x * 16, j = blockIdx.y;
  int tid = threadIdx.x;
  for (int t = tid; t < 64 * 32; t += 256) {
    int m = t >> 5, y = t & 31;
    bcol[t] = b[((size_t)m * 256 + j) * 32 + y];
  }
  if (tid < 16) {
    float s = 0.f;
    for (int m = 0; m < 64; m++) s += mask[m * 256 + i0 + tid] * mask[m * 256 + j];
    nrm[tid] = s;
  }
  int wave = tid >> 5, lane = tid & 31, lr = lane & 15, lh = lane >> 4;
  v8f z = {0.f, 0.f, 0.f, 0.f, 0.f, 0.f, 0.f, 0.f};
  v8f acc = z;
  for (int half = 0; half < 2; half++) {
    __syncthreads();
    // stage 1: P[i][ (x-16*half)*32 + y ] = sum_m a[m,i0+i,x]*bcol[m][y]
#pragma unroll
    for (int xi = 0; xi < 2; xi++) {
      int x = half * 16 + wave + xi * 8;
      v8f c0 = z, c1 = z;
      for (int m0 = 0; m0 < 64; m0 += 4) {
        int ma = m0 + lh * 2;
        v2f av;
        av.x = a[((size_t)ma * 256 + i0 + lr) * 32 + x];
        av.y = a[((size_t)(ma + 1) * 256 + i0 + lr) * 32 + x];
        v2f b0; b0.x = bcol[ma * 32 + lr]; b0.y = bcol[(ma + 1) * 32 + lr];
        v2f b1; b1.x = bcol[ma * 32 + 16 + lr]; b1.y = bcol[(ma + 1) * 32 + 16 + lr];
        c0 = wmma4(av, b0, c0);
        c1 = wmma4(av, b1, c1);
      }
      int xl = x - half * 16;
#pragma unroll
      for (int r = 0; r < 8; r++) {
        int i = r + lh * 8;
        P[i * 512 + xl * 32 + lr] = c0[r];
        P[i * 512 + xl * 32 + 16 + lr] = c1[r];
      }
    }
    __syncthreads();
    // stage 2: partial GEMM over this half's K range; wave -> 16-col chunk
    int cn = wave * 16;
    for (int k0 = 0; k0 < 512; k0 += 4) {
      int ka = k0 + lh * 2;
      v2f av = ldv2(&P[lr * 512 + ka]);
      v2f bv;
      bv.x = Wf[(size_t)(half * 512 + ka) * 128 + cn + lr];
      bv.y = Wf[(size_t)(half * 512 + ka + 1) * 128 + cn + lr];
      acc = wmma4(av, bv, acc);
    }
  }
#pragma unroll
  for (int r = 0; r < 8; r++) {
    int i = r + lh * 8, col = wave * 16 + lr;
    float v = (acc[r] + bf[col]) / (nrm[i] + 1e-3f);
    pair[(((size_t)(i0 + i)) * 256 + j) * 128 + col] += v;
  }
}

// ---------------------------------------------------------------------------
// Triangle multiply einsum over channel-major operands TI/TJ: [128][256][256]
// outgoing: out[i,j,c] = sum_k TI[c][i][k]*TJ[c][j][k]
// ingoing : out[i,j,c] = sum_k TI[c][k][j]*TJ[c][k][i]
// ---------------------------------------------------------------------------
__global__ __launch_bounds__(256) void trimul_kernel(
    const float* __restrict__ TI, const float* __restrict__ TJ,
    float* __restrict__ out, int ingoing) {
  int wave = threadIdx.x >> 5, lane = threadIdx.x & 31;
  int lr = lane & 15, lh = lane >> 4;
  int c = blockIdx.z;
  int j0 = blockIdx.x * 16;
  int i0 = (blockIdx.y * 8 + wave) * 16;
  const float* I = TI + (size_t)c * 65536;
  const float* J = TJ + (size_t)c * 65536;
  v8f acc = {0.f, 0.f, 0.f, 0.f, 0.f, 0.f, 0.f, 0.f};
  if (!ingoing) {
    const float* ip = I + (i0 + lr) * 256 + lh * 2;
    const float* jp = J + (j0 + lr) * 256 + lh * 2;
    for (int k0 = 0; k0 < 256; k0 += 4) {
      v2f a = ldv2(ip + k0);
      v2f b = ldv2(jp + k0);
      acc = wmma4(a, b, acc);
    }
  } else {
    for (int k0 = 0; k0 < 256; k0 += 4) {
      int ka = k0 + lh * 2;
      v2f a, b;
      a.x = J[ka * 256 + i0 + lr]; a.y = J[(ka + 1) * 256 + i0 + lr];
      b.x = I[ka * 256 + j0 + lr]; b.y = I[(ka + 1) * 256 + j0 + lr];
      acc = wmma4(a, b, acc);
    }
  }
#pragma unroll
  for (int r = 0; r < 8; r++) {
    int i = i0 + r + lh * 8, jj = j0 + lr;
    out[((size_t)i * 256 + jj) * 128 + c] = acc[r];
  }
}

// gate+transpose: T[c][r] = lin[r,c] * sigmoid(lin_s[r,c]); R=65536, C=128
__global__ __launch_bounds__(1024) void gate_transpose(
    const float* __restrict__ lin, const float* __restrict__ lin_s,
    float* __restrict__ T, int R, int C) {
  __shared__ float tile[32][33];
  int r0 = blockIdx.x * 32, c0 = blockIdx.y * 32;
  int tx = threadIdx.x, ty = threadIdx.y;
  size_t ii = (size_t)(r0 + ty) * C + c0 + tx;
  tile[ty][tx] = lin[ii] * sigm(lin_s[ii]);
  __syncthreads();
  T[(size_t)(c0 + ty) * R + r0 + tx] = tile[tx][ty];
}

// row permute: out[(b*A + a)*C + c] = in[(a*B + b)*C + c]
__global__ __launch_bounds__(256) void permute_rows(
    const float* __restrict__ in, float* __restrict__ out, int A, int B, int C) {
  int row = blockIdx.x;
  int a = row / B, b = row % B;
  const float* ip = in + (size_t)row * C;
  float* op = out + ((size_t)b * A + a) * C;
  for (int c = threadIdx.x; c < C; c += 256) op[c] = ip[c];
}

// mask biases: kbrow[m*256+i] = 1e4*(mask-1); kbcol[i*64+m] = 1e9*(mask-1)
__global__ __launch_bounds__(256) void make_kb(
    const float* __restrict__ mask, float* __restrict__ kbrow,
    float* __restrict__ kbcol) {
  int idx = blockIdx.x * 256 + threadIdx.x;
  int m = idx >> 8, i = idx & 255;
  float mv = mask[idx];
  kbrow[idx] = 10000.f * (mv - 1.f);
  kbcol[i * 64 + m] = 1e9f * (mv - 1.f);
}

// ---------------------------------------------------------------------------
// Host orchestration.
// Param leaf indices assume recursive dict-insertion-order flattening of
// setup_inputs(): [r1d, pair, mask, row(12), col(9), msa_trans(6), opm(8),
// tm_out(16), tm_in(16), ta_start(10), ta_end(10), pair_trans(6)] = 96 inputs.
// Workspace requirement: ~204 MB (50,888,704 floats).
// ---------------------------------------------------------------------------
extern "C" void kernel_launch(void* const* d_in, const int* in_sizes, int n_in,
                              void* d_out, int out_size, void* d_ws,
                              size_t ws_size, hipStream_t stream) {
  (void)in_sizes; (void)n_in; (void)out_size; (void)ws_size;
  auto in = [&](int i) { return (const float*)d_in[i]; };
  const float* mask = in(2);

  const size_t S1 = (size_t)64 * 256 * 256;   // 4,194,304
  const size_t S2 = (size_t)256 * 256 * 128;  // 8,388,608
  float* r1dO = (float*)d_out;
  float* pairO = r1dO + S1;

  float* pool = (float*)d_ws;
  float* lnA = pool;            // phase A slots (16 MB each)
  float* bq = pool + 1 * S1;
  float* bk = pool + 2 * S1;
  float* bv = pool + 3 * S1;
  float* bg = pool + 4 * S1;
  float* bo = pool + 5 * S1;
  float* t1 = pool;             // phase B slots (32 MB each, overlap phase A)
  float* t2 = pool + 1 * S2;
  float* t3 = pool + 2 * S2;
  float* t4 = pool + 3 * S2;
  float* t5 = pool + 4 * S2;
  float* ln2d = pool + 5 * S2;
  float* pb = ln2d + S2;                    // up to [8][65536]
  float* kbrow = pb + (size_t)8 * 65536;
  float* kbcol = kbrow + 16384;
  const float SC = 0.17677669529663687f;    // 1/sqrt(32)

  hipMemcpyAsync(r1dO, in(0), S1 * sizeof(float), hipMemcpyDeviceToDevice, stream);
  hipMemcpyAsync(pairO, in(1), S2 * sizeof(float), hipMemcpyDeviceToDevice, stream);
  make_kb<<<64, 256, 0, stream>>>(mask, kbrow, kbcol);

  auto LN = [&](const float* x, int gi, int bi, float* o, int R, int C) {
    ln_kernel<<<dim3((R + 7) / 8), 256, 0, stream>>>(x, in(gi), in(bi), o, R, C);
  };
  auto GEMM = [&](const float* A, const float* W, const float* bias, float* o,
                  const float* aux, int M, int N, int K, int flags, int pa,
                  int pbm) {
    gemm_wmma<<<dim3(M / 128, (N + 63) / 64), 256, 0, stream>>>(
        A, W, bias, o, aux, M, N, K, flags, pa, pbm);
  };

  // ---- row attention with pair bias --------------------------------------
  LN(r1dO, 3, 4, lnA, 16384, 256);
  LN(pairO, 12, 13, ln2d, 65536, 128);
  gemm_small_n_t<<<dim3(256, 8), 256, 0, stream>>>(ln2d, in(14), pb, 65536, 8, 128);
  GEMM(lnA, in(5), nullptr, bq, nullptr, 16384, 256, 256, 0, 0, 0);
  GEMM(lnA, in(6), nullptr, bk, nullptr, 16384, 256, 256, 0, 0, 0);
  GEMM(lnA, in(7), nullptr, bv, nullptr, 16384, 256, 256, 0, 0, 0);
  GEMM(lnA, in(8), in(9), bg, nullptr, 16384, 256, 256, 0, 0, 0);
  attn_kernel<<<dim3(64, 8), 256, 0, stream>>>(bq, bk, bv, bg, bo, pb, kbrow,
                                               256, 8, SC);
  GEMM(bo, in(10), in(11), r1dO, nullptr, 16384, 256, 256, FLAG_ACC, 0, 0);

  // ---- column attention ---------------------------------------------------
  LN(r1dO, 15, 16, lnA, 16384, 256);
  permute_rows<<<16384, 256, 0, stream>>>(lnA, bo, 64, 256, 256);  // [m,i]->[i,m]
  GEMM(bo, in(17), nullptr, bq, nullptr, 16384, 256, 256, 0, 0, 0);
  GEMM(bo, in(18), nullptr, bk, nullptr, 16384, 256, 256, 0, 0, 0);
  GEMM(bo, in(19), nullptr, bv, nullptr, 16384, 256, 256, 0, 0, 0);
  GEMM(bo, in(20), in(21), bg, nullptr, 16384, 256, 256, 0, 0, 0);
  attn_kernel<<<dim3(256, 8), 256, 0, stream>>>(bq, bk, bv, bg, lnA, nullptr,
                                                kbcol, 64, 8, SC);
  GEMM(lnA, in(22), in(23), r1dO, nullptr, 16384, 256, 256,
       FLAG_ACC | FLAG_ROWSWAP, 256, 64);

  // ---- MSA transition -----------------------------------------------------
  LN(r1dO, 24, 25, lnA, 16384, 256);
  transition_kernel<256, 1024><<<1024, 256, 0, stream>>>(lnA, in(26), in(27),
                                                         in(28), in(29), r1dO);

  // ---- outer product mean -------------------------------------------------
  LN(r1dO, 30, 31, lnA, 16384, 256);
  GEMM(lnA, in(32), in(33), bq, mask, 16384, 32, 256, FLAG_ROWMASK, 0, 0);
  GEMM(lnA, in(34), in(35), bk, mask, 16384, 32, 256, FLAG_ROWMASK, 0, 0);
  opm_kernel<<<dim3(16, 256), 256, 0, stream>>>(bq, bk, in(36), in(37), mask,
                                                pairO);

  // ---- triangle multiplies ------------------------------------------------
  auto TM = [&](int p, int ingoing) {
    // leaf order: n1g n1b n2g n2b l1i.w/b l1j.w/b l1i_s.w/b l1j_s.w/b l2.w/b l3s.w/b
    LN(pairO, p + 0, p + 1, ln2d, 65536, 128);
    GEMM(ln2d, in(p + 4), in(p + 5), t1, nullptr, 65536, 128, 128, 0, 0, 0);
    GEMM(ln2d, in(p + 8), in(p + 9), t2, nullptr, 65536, 128, 128, 0, 0, 0);
    gate_transpose<<<dim3(2048, 4), dim3(32, 32), 0, stream>>>(t1, t2, t3,
                                                               65536, 128);
    GEMM(ln2d, in(p + 6), in(p + 7), t1, nullptr, 65536, 128, 128, 0, 0, 0);
    GEMM(ln2d, in(p + 10), in(p + 11), t2, nullptr, 65536, 128, 128, 0, 0, 0);
    gate_transpose<<<dim3(2048, 4), dim3(32, 32), 0, stream>>>(t1, t2, t4,
                                                               65536, 128);
    trimul_kernel<<<dim3(16, 2, 128), 256, 0, stream>>>(t3, t4, t1, ingoing);
    LN(t1, p + 2, p + 3, t2, 65536, 128);
    GEMM(ln2d, in(p + 14), in(p + 15), t3, nullptr, 65536, 128, 128, 0, 0, 0);
    GEMM(t2, in(p + 12), in(p + 13), pairO, t3, 65536, 128, 128,
         FLAG_ACC | FLAG_MULSIG, 0, 0);
  };
  TM(38, 0);  // tm_out (outgoing)
  TM(54, 1);  // tm_in  (ingoing)

  // ---- triangle attentions ------------------------------------------------
  auto TA = [&](int p, int transposed) {
    // leaf order: ng nb qw kw vw gw gb fw fb biasw
    LN(pairO, p + 0, p + 1, ln2d, 65536, 128);
    const float* X = ln2d;
    float* aout = t5;
    if (transposed) {
      permute_rows<<<65536, 256, 0, stream>>>(ln2d, t5, 256, 256, 128);
      X = t5;
      aout = ln2d;
    }
    gemm_small_n_t<<<dim3(256, 4), 256, 0, stream>>>(X, in(p + 9), pb, 65536, 4,
                                                     128);
    GEMM(X, in(p + 2), nullptr, t1, nullptr, 65536, 128, 128, 0, 0, 0);
    GEMM(X, in(p + 3), nullptr, t2, nullptr, 65536, 128, 128, 0, 0, 0);
    GEMM(X, in(p + 4), nullptr, t3, nullptr, 65536, 128, 128, 0, 0, 0);
    GEMM(X, in(p + 5), in(p + 6), t4, nullptr, 65536, 128, 128, 0, 0, 0);
    attn_kernel<<<dim3(256, 4), 256, 0, stream>>>(t1, t2, t3, t4, aout, pb,
                                                  nullptr, 256, 4, SC);
    GEMM(aout, in(p + 7), in(p + 8), pairO, nullptr, 65536, 128, 128,
         FLAG_ACC | (transposed ? FLAG_ROWSWAP : 0), 256, 256);
  };
  TA(70, 0);  // ta_start
  TA(80, 1);  // ta_end

  // ---- pair transition ----------------------------------------------------
  LN(pairO, 90, 91, ln2d, 65536, 128);
  transition_kernel<128, 512><<<4096, 256, 0, stream>>>(ln2d, in(92), in(93),
                                                        in(94), in(95), pairO);
}